// TimeIntervalAwareSelfAttention_78700980732208
// MI455X (gfx1250) — compile-verified
//
#include <hip/hip_runtime.h>
#include <hip/hip_bf16.h>
#include <stdint.h>

// Problem constants from the reference
#define BB 8
#define LL 256
#define HID 128
#define NH 4
#define DD 32
#define MROWS (BB * LL)  // 2048

typedef __attribute__((ext_vector_type(2))) float v2f;
typedef __attribute__((ext_vector_type(8))) float v8f;

static __device__ __forceinline__ v8f vzero8() {
    v8f z = {0.f, 0.f, 0.f, 0.f, 0.f, 0.f, 0.f, 0.f};
    return z;
}

// ---------------- f32 WMMA (V_WMMA_F32_16X16X4_F32) ----------------
// A frag (ISA 7.12.2): lane<16 -> (A[m][kb+0], A[m][kb+1]); lane>=16 -> (A[m][kb+2], A[m][kb+3]).
// B frag:              lane<16 -> (B[kb+0][n], B[kb+1][n]);  lane>=16 -> (B[kb+2][n], B[kb+3][n]).
// C/D: vgpr v -> element (m = v + (lane>=16 ? 8:0), n = lane&15).
static __device__ __forceinline__ v8f wmma_f32_4(v2f a, v2f b, v8f c) {
#if __has_builtin(__builtin_amdgcn_wmma_f32_16x16x4_f32)
    return __builtin_amdgcn_wmma_f32_16x16x4_f32(false, a, false, b, (short)0, c, false, false);
#else
    int l = threadIdx.x & 31;
    float ax = a.x, ay = a.y, bx = b.x, by = b.y;
#pragma unroll
    for (int v = 0; v < 8; ++v) {
        int m = v + ((l & 16) ? 8 : 0);
        int n = l & 15;
        float a0 = __shfl(ax, m, 32), a1 = __shfl(ay, m, 32);
        float a2 = __shfl(ax, m + 16, 32), a3 = __shfl(ay, m + 16, 32);
        float b0 = __shfl(bx, n, 32), b1 = __shfl(by, n, 32);
        float b2 = __shfl(bx, n + 16, 32), b3 = __shfl(by, n + 16, 32);
        c[v] += a0 * b0 + a1 * b1 + a2 * b2 + a3 * b3;
    }
    return c;
#endif
}

// ------------- gfx1250 async global->LDS copy (ASYNCcnt path) -------------
// Builtin signature (from hipcc diagnostic): param0 = 'vector_size(16) int __device__*'
// i.e. int4 in addrspace(1); param1 = int4 in addrspace(3); then imm offset, imm cpol.
#if __has_builtin(__builtin_amdgcn_global_load_async_to_lds_b128)
#define HAS_ASYNC_LDS 1
typedef int i4_t __attribute__((vector_size(16)));
typedef __attribute__((address_space(1))) i4_t* gp4_t;  // pointer to global int4
typedef __attribute__((address_space(3))) i4_t* lp4_t;  // pointer to LDS int4
#else
#define HAS_ASYNC_LDS 0
#endif

static __device__ __forceinline__ void cp_f4(float* ldst, const float* gsrc) {
#if HAS_ASYNC_LDS
    __builtin_amdgcn_global_load_async_to_lds_b128(
        (gp4_t)(uintptr_t)gsrc, (lp4_t)(uint32_t)(uintptr_t)ldst, 0, 0);
#else
    *(float4*)ldst = *(const float4*)gsrc;
#endif
}

static __device__ __forceinline__ void cp_commit_wait() {
#if HAS_ASYNC_LDS
#if __has_builtin(__builtin_amdgcn_s_wait_asynccnt)
    __builtin_amdgcn_s_wait_asynccnt(0);
#else
    asm volatile("s_wait_asynccnt 0" ::: "memory");
#endif
#endif
}

// out[row,n] = sum_k A[row,k]*W[k,n] + bias[n] (+ Epos[pos[row],n]) ; optional NaN->0
__global__ void __launch_bounds__(256)
gemm_kernel(const float* __restrict__ A, const float* __restrict__ W,
            const float* __restrict__ bias, const float* __restrict__ Epos,
            const int* __restrict__ pos, float* __restrict__ out, int nan_clean) {
    __shared__ __align__(16) float as[16 * HID];  // 8 KB, contiguous strip of A

    int row0 = blockIdx.x * 16;
    // 16x128 strip is contiguous in global: 512 float4's, async-copied to LDS.
    for (int i = threadIdx.x; i < (16 * HID) / 4; i += 256)
        cp_f4(as + i * 4, A + row0 * HID + i * 4);
    cp_commit_wait();
    __syncthreads();

    int l = threadIdx.x & 31;
    int wv = threadIdx.x >> 5;
    int n0 = wv * 16;

    int mrow = l & 15;
    int khi = (l & 16) ? 2 : 0;
    int col = n0 + (l & 15);

    v8f acc = vzero8();
#pragma unroll 4
    for (int kb = 0; kb < HID; kb += 4) {
        int k0 = kb + khi;
        v2f a;
        a.x = as[mrow * HID + k0];
        a.y = as[mrow * HID + k0 + 1];
        v2f bf;
        bf.x = W[k0 * HID + col];
        bf.y = W[(k0 + 1) * HID + col];
        acc = wmma_f32_4(a, bf, acc);
    }

#pragma unroll
    for (int v = 0; v < 8; ++v) {
        int m = v + ((l & 16) ? 8 : 0);
        int row = row0 + m;
        float val = acc[v] + bias[col];
        if (Epos) val += Epos[pos[row] * HID + col];
        if (nan_clean && !(val == val)) val = 0.f;
        out[row * HID + col] = val;
    }
}

// Fused attention per (b, h, 16-query tile).
// LDS (dynamic, all 16B-aligned sections):
//   erk[257*32] | erv[257*32] | qtile[16*32] | wbuf[16*260] | idx[16*256] (int) | obuf[16*32]
#define ERK_F (257 * DD)
#define WB_STRIDE 260
#define ATTN_LDS_FLOATS (ERK_F + ERK_F + 16 * DD + 16 * WB_STRIDE + 16 * LL + 16 * DD)

__global__ void __launch_bounds__(256)
attn_kernel(const float* __restrict__ Qb, const float* __restrict__ Kb,
            const float* __restrict__ Vb, const int* __restrict__ intv,
            const unsigned char* __restrict__ amask,
            const float* __restrict__ E_RK, const float* __restrict__ E_RV,
            float* __restrict__ Ab) {
    extern __shared__ __align__(16) float smem[];
    float* erk = smem;                          // 257*32
    float* erv = erk + ERK_F;                   // 257*32
    float* qtile = erv + ERK_F;                 // 16*32
    float* wbuf = qtile + 16 * DD;              // 16*260
    int* idxb = (int*)(wbuf + 16 * WB_STRIDE);  // 16*256
    float* obuf = (float*)(idxb + 16 * LL);     // 16*32

    int x = blockIdx.x;
    int qt = x & 15;       // query tile
    int h = (x >> 4) & 3;  // head
    int b = x >> 6;        // batch

    int t = threadIdx.x;
    int l = t & 31;
    int wv = t >> 5;

    // ---- async-stage everything into LDS (E tables kill the 2x268MB rel materialization) ----
    // rel tables, head slice: 257 rows x 32 floats (8 float4 per row, strided source)
    for (int i = t; i < 257 * (DD / 4); i += 256) {  // 2056 float4
        int r = i >> 3;
        int c4 = i & 7;
        cp_f4(erk + i * 4, E_RK + r * HID + h * DD + c4 * 4);
        cp_f4(erv + i * 4, E_RV + r * HID + h * DD + c4 * 4);
    }
    // Q tile: 16 rows x 8 float4 (strided source)
    for (int i = t; i < 16 * (DD / 4); i += 256) {
        int m = i >> 3;
        int c4 = i & 7;
        cp_f4(qtile + i * 4, Qb + (b * LL + qt * 16 + m) * HID + h * DD + c4 * 4);
    }
    // interval indices: 16x256 block is fully contiguous in global (16KB)
    {
        const float* src = (const float*)(intv + (b * LL + qt * 16) * LL);
        for (int i = t; i < (16 * LL) / 4; i += 256)
            cp_f4((float*)(idxb + i * 4), src + i * 4);
    }
    cp_commit_wait();
    __syncthreads();

    // ---- scores: Qh.Kh^T via WMMA + rel_k gather (from LDS), masked, scaled ----
    const float scale = 0.17677669529663687f;  // 1/sqrt(32)
    int mrow = l & 15;
    int khi = (l & 16) ? 2 : 0;
#pragma unroll
    for (int kti = 0; kti < 2; ++kti) {
        int kt = wv + kti * 8;  // 16 key tiles over 8 waves
        int key_col = kt * 16 + (l & 15);
        v8f acc = vzero8();
#pragma unroll
        for (int db = 0; db < DD; db += 4) {
            int k0 = db + khi;
            v2f a;
            a.x = qtile[mrow * DD + k0];
            a.y = qtile[mrow * DD + k0 + 1];
            v2f bf;  // B[d][key] = Kh[key][d]
            const float* Kp = Kb + (b * LL + key_col) * HID + h * DD + k0;
            bf.x = Kp[0];
            bf.y = Kp[1];
            acc = wmma_f32_4(a, bf, acc);
        }
#pragma unroll
        for (int v = 0; v < 8; ++v) {
            int m = v + ((l & 16) ? 8 : 0);
            int n = l & 15;
            int key = kt * 16 + n;
            int q = qt * 16 + m;
            const float* er = erk + idxb[m * LL + key] * DD;
            const float* qr = qtile + m * DD;
            float rel = 0.f;
#pragma unroll
            for (int d = 0; d < DD; ++d) rel += qr[d] * er[d];
            float s = (acc[v] + rel) * scale;
            unsigned char mk = amask[(size_t)b * LL * LL + (size_t)q * LL + key];
            wbuf[m * WB_STRIDE + key] = mk ? s : -__builtin_inff();
        }
    }
    __syncthreads();

    // zero the output accumulator (merged via ds_add_f32 atomics below)
    obuf[t] = 0.f;
    obuf[t + 256] = 0.f;

    // ---- softmax: 16 lanes per row, shuffle-tree reductions (wave32, xor<16 stays in row group) ----
    {
        int m = t >> 4;
        int sl = t & 15;
        float* row = wbuf + m * WB_STRIDE;
        float mx = -__builtin_inff();
        for (int k = sl; k < LL; k += 16) mx = fmaxf(mx, row[k]);
#pragma unroll
        for (int off = 8; off; off >>= 1) mx = fmaxf(mx, __shfl_xor(mx, off, 32));
        if (!(mx > -__builtin_inff())) {
            for (int k = sl; k < LL; k += 16) row[k] = 0.f;  // fully-masked row
        } else {
            float s = 0.f;
            for (int k = sl; k < LL; k += 16) {
                float e = __expf(row[k] - mx);
                row[k] = e;
                s += e;
            }
#pragma unroll
            for (int off = 8; off; off >>= 1) s += __shfl_xor(s, off, 32);
            float inv = 1.f / s;
            for (int k = sl; k < LL; k += 16) row[k] *= inv;
        }
    }
    __syncthreads();

    // ---- out = w @ Vh via WMMA: all 8 waves busy (2 d-tiles x 4 K-slices of 64) ----
    {
        int nt = wv >> 2;  // d tile 0..1
        int ks = wv & 3;   // K slice 0..3
        int n0 = nt * 16;
        int dcol = n0 + (l & 15);
        v8f acc = vzero8();
#pragma unroll 4
        for (int kb = ks * 64; kb < ks * 64 + 64; kb += 4) {
            int k0 = kb + khi;
            v2f a;
            a.x = wbuf[mrow * WB_STRIDE + k0];
            a.y = wbuf[mrow * WB_STRIDE + k0 + 1];
            v2f bf;  // B[k][d] = Vh[k][d]
            const float* Vp = Vb + (b * LL + k0) * HID + h * DD + dcol;
            bf.x = Vp[0];
            bf.y = Vp[HID];  // k0+1
            acc = wmma_f32_4(a, bf, acc);
        }
#pragma unroll
        for (int v = 0; v < 8; ++v) {
            int m = v + ((l & 16) ? 8 : 0);
            atomicAdd(&obuf[m * DD + n0 + (l & 15)], acc[v]);  // ds_add_f32
        }
    }

    // ---- rel_v term: each thread owns two (m,d) outputs, gathered from LDS ----
    {
        float rel0 = 0.f, rel1 = 0.f;
        int m0 = t >> 5, d0 = t & 31;
        int m1 = (t + 256) >> 5;
        for (int k = 0; k < LL; ++k) {
            rel0 += wbuf[m0 * WB_STRIDE + k] * erv[idxb[m0 * LL + k] * DD + d0];
            rel1 += wbuf[m1 * WB_STRIDE + k] * erv[idxb[m1 * LL + k] * DD + d0];
        }
        atomicAdd(&obuf[t], rel0);
        atomicAdd(&obuf[t + 256], rel1);
    }
    __syncthreads();

    // ---- write (b, q, h*32+d) ----
    {
        int p = t;
#pragma unroll
        for (int i = 0; i < 2; ++i, p += 256) {
            int m = p >> 5, d = p & 31;
            Ab[(b * LL + qt * 16 + m) * HID + h * DD + d] = obuf[p];
        }
    }
}

extern "C" void kernel_launch(void* const* d_in, const int* in_sizes, int n_in,
                              void* d_out, int out_size, void* d_ws, size_t ws_size,
                              hipStream_t stream) {
    (void)in_sizes; (void)n_in; (void)out_size; (void)ws_size;
    const float* query = (const float*)d_in[0];
    const float* key   = (const float*)d_in[1];
    const float* value = (const float*)d_in[2];
    const int* poss    = (const int*)d_in[3];
    const int* intv    = (const int*)d_in[4];
    const unsigned char* amask = (const unsigned char*)d_in[5];
    const float* Wq = (const float*)d_in[6];
    const float* bq = (const float*)d_in[7];
    const float* Wk = (const float*)d_in[8];
    const float* bk = (const float*)d_in[9];
    const float* Wv = (const float*)d_in[10];
    const float* bv = (const float*)d_in[11];
    const float* Wo = (const float*)d_in[12];
    const float* bo = (const float*)d_in[13];
    const float* E_PK = (const float*)d_in[14];
    const float* E_PV = (const float*)d_in[15];
    const float* E_RK = (const float*)d_in[16];
    const float* E_RV = (const float*)d_in[17];
    float* out = (float*)d_out;

    float* Qb = (float*)d_ws;  // 2048*128 each
    float* Kb = Qb + MROWS * HID;
    float* Vb = Kb + MROWS * HID;
    float* Ab = Vb + MROWS * HID;  // 4 MB total workspace

    dim3 blk(256);
    gemm_kernel<<<MROWS / 16, blk, 0, stream>>>(query, Wq, bq, nullptr, nullptr, Qb, 0);
    gemm_kernel<<<MROWS / 16, blk, 0, stream>>>(key,   Wk, bk, E_PK, poss, Kb, 0);
    gemm_kernel<<<MROWS / 16, blk, 0, stream>>>(value, Wv, bv, E_PV, poss, Vb, 0);

    attn_kernel<<<BB * NH * (LL / 16), blk, ATTN_LDS_FLOATS * sizeof(float), stream>>>(
        Qb, Kb, Vb, intv, amask, E_RK, E_RV, Ab);

    gemm_kernel<<<MROWS / 16, blk, 0, stream>>>(Ab, Wo, bo, nullptr, nullptr, out, 1);
}